// TopologyBatch_41755672051932
// MI455X (gfx1250) — compile-verified
//
#include <hip/hip_runtime.h>
#include <math.h>

typedef __attribute__((ext_vector_type(2))) float v2f;
typedef __attribute__((ext_vector_type(8))) float v8f;

#define B_ 64
#define F_ 240
#define J_ 22
#define NWAVES 8

__constant__ int d_SEG_S[16] = {2,5,8, 1,4,7, 3,6,9,12, 14,17,19, 13,16,18};
__constant__ int d_SEG_E[16] = {5,8,11, 4,7,10, 6,9,12,15, 17,19,21, 16,18,20};
__constant__ int d_PATH[16]  = {0,0,0, 1,1,1, 2,2,2,2, 3,3,3, 4,4,4};

// Wave-local LDS fence: all buffers are per-wave, and LDS ops from one wave are
// processed in order by the LDS pipe (DScnt), so a dscnt==0 wait is sufficient —
// no block-wide s_barrier needed.
#define WAVE_LDS_FENCE() asm volatile("s_wait_dscnt 0x0" ::: "memory")

struct F3 { float x, y, z; };
__device__ __forceinline__ F3 f3sub(F3 a, F3 b) { return {a.x-b.x, a.y-b.y, a.z-b.z}; }
__device__ __forceinline__ F3 f3cross(F3 a, F3 b) {
    return {a.y*b.z - a.z*b.y, a.z*b.x - a.x*b.z, a.x*b.y - a.y*b.x};
}
__device__ __forceinline__ float f3dot(F3 a, F3 b) { return a.x*b.x + a.y*b.y + a.z*b.z; }
__device__ __forceinline__ F3 f3norm(F3 v) {
    float ss = f3dot(v, v);
    float inv = (ss > 0.f) ? rsqrtf(ss) : 0.f;
    return {v.x*inv, v.y*inv, v.z*inv};
}
__device__ __forceinline__ float asindot(F3 a, F3 b) {
    float d = f3dot(a, b);
    d = fminf(fmaxf(d, -1.f), 1.f);
    return asinf(d);
}
__device__ __forceinline__ float gli_pair(F3 s1, F3 e1, F3 s2, F3 e2) {
    F3 r13 = f3sub(s2, s1), r14 = f3sub(e2, s1);
    F3 r23 = f3sub(s2, e1), r24 = f3sub(e2, e1);
    F3 r12 = f3sub(e1, s1), r34 = f3sub(e2, s2);
    F3 f0 = f3norm(f3cross(r13, r14));
    F3 f1 = f3norm(f3cross(r14, r24));
    F3 f2 = f3norm(f3cross(r24, r23));
    F3 f3v = f3norm(f3cross(r23, r13));
    float total = asindot(f0, f1) + asindot(f1, f2) + asindot(f2, f3v) + asindot(f3v, f0);
    float sgn = f3dot(f3cross(r34, r12), r13);
    float mult = (sgn <= 0.f) ? -1.f : 1.f;
    return mult * total * 0.07957747154594767f;  // 1/(4*pi)
}
__device__ __forceinline__ F3 jget(const float* jb, int j) {
    return {jb[j*3 + 0], jb[j*3 + 1], jb[j*3 + 2]};
}

// ---------------- Kernel A: per-frame xz bbox overlap flags ----------------
__global__ void flags_kernel(const float* __restrict__ m1, const float* __restrict__ m2,
                             int* __restrict__ flags) {
    int t = blockIdx.x * blockDim.x + threadIdx.x;
    if (t >= B_ * F_) return;
    const float* p1 = m1 + (size_t)t * (J_ * 3);
    const float* p2 = m2 + (size_t)t * (J_ * 3);
    float nx1 = p1[0], xx1 = p1[0], nz1 = p1[2], xz1 = p1[2];
    float nx2 = p2[0], xx2 = p2[0], nz2 = p2[2], xz2 = p2[2];
    for (int j = 1; j < J_; ++j) {
        float x1 = p1[j*3], z1 = p1[j*3 + 2];
        float x2 = p2[j*3], z2 = p2[j*3 + 2];
        nx1 = fminf(nx1, x1); xx1 = fmaxf(xx1, x1);
        nz1 = fminf(nz1, z1); xz1 = fmaxf(xz1, z1);
        nx2 = fminf(nx2, x2); xx2 = fmaxf(xx2, x2);
        nz2 = fminf(nz2, z2); xz2 = fmaxf(xz2, z2);
    }
    bool fl = (xx1 >= nx2) && (xx2 >= nx1) && (xz1 >= nz2) && (xz2 >= nz1);
    flags[t] = fl ? 1 : 0;
}

// Per-frame pipeline: stage joints -> GLI (C-fragment layout) -> LDS relayout ->
// U = P^T x G -> LDS relayout -> V = P^T x U^T. Returns V fragment (v8f).
__device__ __forceinline__ v8f frame_V(const float* __restrict__ m1,
                                       const float* __restrict__ m2,
                                       size_t base,
                                       float (&jb)[136], float (&gl)[256], float (&ul)[256],
                                       int lane, int n, int hi, int kb,
                                       const v2f (&aop)[4]) {
    // cooperative joint stage: motion1 -> jb[0..65], motion2 -> jb[66..131]
    for (int t = lane; t < 2 * J_ * 3; t += 32) {
        const float* p = (t < J_ * 3) ? (m1 + base + t) : (m2 + base + (t - J_ * 3));
        jb[t] = *p;
    }
    WAVE_LDS_FENCE();

    // G[m,n] = GLI(segment m of motion1, segment n of motion2); lane holds
    // G[(lane>>4)*8 + v, lane&15] == the 16x16 f32 C/D WMMA fragment layout.
    F3 s2 = jget(&jb[66], d_SEG_S[n]);
    F3 e2 = jget(&jb[66], d_SEG_E[n]);
    float g[8];
#pragma unroll
    for (int v = 0; v < 8; ++v) {
        int ms = hi * 8 + v;
        F3 s1 = jget(&jb[0], d_SEG_S[ms]);
        F3 e1 = jget(&jb[0], d_SEG_E[ms]);
        g[v] = gli_pair(s1, e1, s2, e2);
    }
    // store G column-major (contiguous in m -> two ds_store_b128 per lane)
    *reinterpret_cast<float4*>(&gl[n*16 + hi*8 + 0]) = make_float4(g[0], g[1], g[2], g[3]);
    *reinterpret_cast<float4*>(&gl[n*16 + hi*8 + 4]) = make_float4(g[4], g[5], g[6], g[7]);
    WAVE_LDS_FENCE();

    // U = P^T x G : K=16 via 4 chained V_WMMA_F32_16X16X4_F32
    v8f u = {0.f, 0.f, 0.f, 0.f, 0.f, 0.f, 0.f, 0.f};
#pragma unroll
    for (int c = 0; c < 4; ++c) {
        float2 bp = *reinterpret_cast<const float2*>(&gl[n*16 + 4*c + kb]);
        v2f bop; bop[0] = bp.x; bop[1] = bp.y;  // B[K=4c+kb, n], B[K=4c+kb+1, n]
        u = __builtin_amdgcn_wmma_f32_16x16x4_f32(
                false, aop[c], false, bop, (short)0, u, false, false);
    }
    // store U row-major: lane's u[v] = U[hi*8+v, n]
#pragma unroll
    for (int v = 0; v < 8; ++v) ul[(hi*8 + v)*16 + n] = u[v];
    WAVE_LDS_FENCE();

    // V = P^T x U^T  (= D^T; transpose is irrelevant for the final max|.|)
    v8f vv = {0.f, 0.f, 0.f, 0.f, 0.f, 0.f, 0.f, 0.f};
#pragma unroll
    for (int c = 0; c < 4; ++c) {
        float2 bp = *reinterpret_cast<const float2*>(&ul[n*16 + 4*c + kb]);
        v2f bop; bop[0] = bp.x; bop[1] = bp.y;  // U^T[K, i] = U[i, K] = ul[i*16 + K]
        vv = __builtin_amdgcn_wmma_f32_16x16x4_f32(
                false, aop[c], false, bop, (short)0, vv, false, false);
    }
    return vv;
}

// ---------------- Kernel B: one wave32 per output (b,f) ----------------
__global__ __launch_bounds__(NWAVES * 32) void gli_kernel(
        const float* __restrict__ m1, const float* __restrict__ m2,
        const int* __restrict__ flags, float* __restrict__ out) {
    __shared__ __align__(16) float jbuf[NWAVES][136];
    __shared__ __align__(16) float glds[NWAVES][256];
    __shared__ __align__(16) float ulds[NWAVES][256];

    const int lane = threadIdx.x & 31;
    const int wv   = threadIdx.x >> 5;
    const int w    = blockIdx.x * NWAVES + wv;    // exact grid: no bounds divergence
    const int b    = w / (F_ - 1);
    const int f    = w % (F_ - 1);

    const int n  = lane & 15;     // segment2 index / result M-row / B-fragment N-col
    const int hi = lane >> 4;
    const int kb = hi * 2;        // f32 A/B fragment: VGPR0 <-> K = 4c+kb, VGPR1 <-> K+1

    // Constant A operand for both matmuls: A_c[M=i, K=k] = P^T[i, 4c+k] = (PATH_ID[4c+k]==i)
    v2f aop[4];
#pragma unroll
    for (int c = 0; c < 4; ++c) {
        aop[c][0] = (d_PATH[4*c + kb]     == n) ? 1.f : 0.f;
        aop[c][1] = (d_PATH[4*c + kb + 1] == n) ? 1.f : 0.f;
    }

    const size_t base0 = ((size_t)b * F_ + f) * (J_ * 3);
    v8f vc = frame_V(m1, m2, base0,            jbuf[wv], glds[wv], ulds[wv], lane, n, hi, kb, aop);
    v8f vn = frame_V(m1, m2, base0 + (J_ * 3), jbuf[wv], glds[wv], ulds[wv], lane, n, hi, kb, aop);

    // dilated bbox masks for frames f and f+1
    const int fi   = b * F_ + f;
    const int flm1 = (f >= 1)     ? flags[fi - 1] : 0;
    const int fl0  = flags[fi];
    const int fl1  = flags[fi + 1];
    const int fl2  = (f + 2 < F_) ? flags[fi + 2] : 0;
    const float mc = (flm1 | fl0 | fl1) ? 1.f : 0.f;
    const float mn = (fl0 | fl1 | fl2)  ? 1.f : 0.f;

    float pm = 0.f;
#pragma unroll
    for (int v = 0; v < 8; ++v)
        pm = fmaxf(pm, fabsf(mn * vn[v] - mc * vc[v]));
#pragma unroll
    for (int off = 16; off > 0; off >>= 1)
        pm = fmaxf(pm, __shfl_xor(pm, off, 32));
    if (lane == 0) out[w] = pm;
}

extern "C" void kernel_launch(void* const* d_in, const int* in_sizes, int n_in,
                              void* d_out, int out_size, void* d_ws, size_t ws_size,
                              hipStream_t stream) {
    const float* m1 = (const float*)d_in[0];
    const float* m2 = (const float*)d_in[1];
    int*   flags = (int*)d_ws;           // B_*F_ ints = 61.4 KB of scratch
    float* out   = (float*)d_out;        // [B, F-1] = 15296 floats

    const int totalF = B_ * F_;
    hipLaunchKernelGGL(flags_kernel, dim3((totalF + 255) / 256), dim3(256), 0, stream,
                       m1, m2, flags);

    const int waves = B_ * (F_ - 1);     // 15296, divisible by NWAVES
    hipLaunchKernelGGL(gli_kernel, dim3(waves / NWAVES), dim3(NWAVES * 32), 0, stream,
                       m1, m2, flags, out);
}